// EFB_22393959481864
// MI455X (gfx1250) — compile-verified
//
#include <hip/hip_runtime.h>

typedef __attribute__((ext_vector_type(2))) float v2f;
typedef __attribute__((ext_vector_type(4))) float v4f;
typedef __attribute__((ext_vector_type(8))) float v8f;
typedef __attribute__((ext_vector_type(4))) unsigned int u32x4;
typedef __attribute__((ext_vector_type(4))) int i32x4;
typedef __attribute__((ext_vector_type(8))) int i32x8;

// workspace layout (in floats)
#define WS_PSUM   0          // 2048
#define WS_PSUMSQ 2048       // 2048
#define WS_AN     4096       // 256  (bn scale)
#define WS_BN     4352       // 256  (bn shift)
#define WS_MU     4608       // 8*9*256 = 18432 (mu_ext, class 8 zeroed)
#define WS_GTAB   23040      // 8*46800 = 374400

// per-batch table: [conv(2)][tap(9)][l(10)][cpad(260)]
#define CPAD        260
#define TAB_PER_B   46800    // 2*9*10*260
#define CONV_STRIDE 23400    // 9*10*260

// ---------------- Kernel 1: per-(b,c) partial BN stats ----------------
__global__ void k_stats(const float* __restrict__ x, float* __restrict__ ws) {
    int c = blockIdx.x & 255;
    int b = blockIdx.x >> 8;
    const v4f* xp = (const v4f*)(x + (((size_t)(b * 256 + c)) << 14));
    float s = 0.f, ss = 0.f;
    for (int i = threadIdx.x; i < 4096; i += 256) {
        v4f v = __builtin_nontemporal_load(xp + i);
        s  += v.x + v.y + v.z + v.w;
        ss += v.x * v.x + v.y * v.y + v.z * v.z + v.w * v.w;
    }
    __shared__ float s1[256], s2[256];
    s1[threadIdx.x] = s; s2[threadIdx.x] = ss;
    __syncthreads();
    for (int off = 128; off > 0; off >>= 1) {
        if ((int)threadIdx.x < off) {
            s1[threadIdx.x] += s1[threadIdx.x + off];
            s2[threadIdx.x] += s2[threadIdx.x + off];
        }
        __syncthreads();
    }
    if (threadIdx.x == 0) {
        ws[WS_PSUM   + blockIdx.x] = s1[0];
        ws[WS_PSUMSQ + blockIdx.x] = s2[0];
    }
}

// ---------------- Kernel 1b: finalize BN scale/shift ----------------
__global__ void k_finstats(const float* __restrict__ bw, const float* __restrict__ bb,
                           float* __restrict__ ws) {
    int c = threadIdx.x;
    float s = 0.f, ss = 0.f;
    for (int b = 0; b < 8; ++b) {
        s  += ws[WS_PSUM   + (b << 8) + c];
        ss += ws[WS_PSUMSQ + (b << 8) + c];
    }
    float mean = s * (1.f / 131072.f);
    float var  = ss * (1.f / 131072.f) - mean * mean;
    float inv  = rsqrtf(var + 1e-5f);
    float a    = bw[c] * inv;
    ws[WS_AN + c] = a;
    ws[WS_BN + c] = bb[c] - a * mean;
}

// ---------------- Kernel 2: per-class FC + ReLU via WMMA ----------------
// mu[b,j,d] = relu( sum_s sel[b,j,s]*fc_w[j,d,s] + fc_b[j,d] )
// GEMM per j: D[256d x 8b] = fc_w[j] (dxs) * sel^T (sxb). 128 waves total.
// Out-of-range columns (n>=8) load clamped pointers unconditionally; their
// results are discarded (WMMA column n depends only on B column n).
__global__ void k_mu(const float* __restrict__ fcw, const float* __restrict__ fcb,
                     const float* __restrict__ style, const int* __restrict__ exist,
                     float* __restrict__ ws) {
    int gw   = blockIdx.x * 4 + (threadIdx.x >> 5);   // 0..127
    int j    = gw >> 4;
    int mt   = gw & 15;
    int lane = threadIdx.x & 31;
    int M    = lane & 15;
    int Kh   = lane >> 4;                             // K0 = 2*Kh
    int d0   = mt << 4;

    const float* Abase = fcw + (((size_t)(j * 256 + d0 + M)) << 8) + (Kh << 1);
    int  n     = M;                                   // output column = batch
    bool valid = n < 8;
    int  nn    = valid ? n : 0;
    int  row   = (exist[nn * 8 + j] == 1) ? j : 8;
    const float* Bbase = style + (((size_t)(nn * 9 + row)) << 8) + (Kh << 1);

    v8f acc = {0.f, 0.f, 0.f, 0.f, 0.f, 0.f, 0.f, 0.f};
    for (int s0 = 0; s0 < 256; s0 += 4) {
        v2f a  = *(const v2f*)(Abase + s0);
        v2f bv = *(const v2f*)(Bbase + s0);
        acc = __builtin_amdgcn_wmma_f32_16x16x4_f32(false, a, false, bv,
                                                    (short)0, acc, false, false);
    }
    float* mu = ws + WS_MU;
#pragma unroll
    for (int v = 0; v < 8; ++v) {
        int   r   = v + (Kh << 3);
        float val = acc[v] + fcb[(j << 8) + d0 + r];
        val = fmaxf(val, 0.f);
        if (valid) {
            mu[((n * 9 + j) << 8) + d0 + r] = val;
            if (j == 0) mu[((n * 9 + 8) << 8) + d0 + r] = 0.f;  // class J -> zeros
        }
    }
}

// ---------------- Kernel 3: conv-weight x mu tables via WMMA ----------------
// v[b,conv,tap,l,c] = sum_s W_conv[c,s,tap] * mu_ext[b,l,s]
// GEMM per (conv,tap): D[256c x 72(b,l)]. 2*9*16*5 = 1440 waves.
__global__ void k_tab(const float* __restrict__ wg, const float* __restrict__ wb,
                      float* __restrict__ ws) {
    int gw   = blockIdx.x * 4 + (threadIdx.x >> 5);   // 0..1439
    int nt   = gw % 5;
    int r    = gw / 5;
    int mt   = r & 15; r >>= 4;
    int tap  = r % 9;
    int conv = r / 9;
    int lane = threadIdx.x & 31;
    int M    = lane & 15;
    int Kh   = lane >> 4;
    int c0   = mt << 4;

    const float* W     = conv ? wb : wg;
    const float* Abase = W + (size_t)(c0 + M) * 2304 + tap + (size_t)(Kh << 1) * 9;
    int  n     = (nt << 4) + M;
    bool valid = n < 72;
    const float* Bbase = ws + WS_MU + (((size_t)(valid ? n : 0)) << 8) + (Kh << 1);

    v8f acc = {0.f, 0.f, 0.f, 0.f, 0.f, 0.f, 0.f, 0.f};
    for (int s0 = 0; s0 < 256; s0 += 4) {
        v2f a;
        a.x = Abase[(size_t)(s0 + 0) * 9];
        a.y = Abase[(size_t)(s0 + 1) * 9];
        v2f bv = *(const v2f*)(Bbase + s0);
        acc = __builtin_amdgcn_wmma_f32_16x16x4_f32(false, a, false, bv,
                                                    (short)0, acc, false, false);
    }
    if (valid) {
        int    b   = n / 9, l = n % 9;
        float* dst = ws + WS_GTAB + (size_t)b * TAB_PER_B
                   + (size_t)((conv * 9 + tap) * 10 + l) * CPAD;
#pragma unroll
        for (int v = 0; v < 8; ++v) {
            int c = c0 + v + (Kh << 3);
            dst[c] = acc[v];
        }
    }
}

// ---------------- Kernel 4: fused gather-conv + BN + affine ----------------
// One TDM descriptor stages the 187 KB per-batch table into LDS.
__global__ __launch_bounds__(512) void k_final(
    const float* __restrict__ x, const int* __restrict__ seg,
    const float* __restrict__ cgb, const float* __restrict__ cbb,
    const float* __restrict__ ws, float* __restrict__ out) {
    extern __shared__ float lds[];
    int b  = blockIdx.x >> 5;
    int rt = blockIdx.x & 31;

    // --- TDM: table (contiguous 46800 floats) -> first 46800 floats of LDS, wave 0 only ---
    if (threadIdx.x == 0) {
        unsigned long long gaddr =
            (unsigned long long)(ws + WS_GTAB + (size_t)b * TAB_PER_B);
        u32x4 g0;
        g0.x = 1u;                                         // count=1 (valid, user)
        g0.y = __builtin_amdgcn_groupstaticsize();         // lds_addr (bytes)
        g0.z = (unsigned int)(gaddr & 0xFFFFFFFFu);        // global_addr[31:0]
        g0.w = (unsigned int)((gaddr >> 32) & 0x01FFFFFFu) // global_addr[56:32]
             | (2u << 30);                                 // type=2 ("image")
        i32x8 g1;
        g1[0] = (int)(2u << 16);                           // data_size=4B; mask=0
        g1[1] = (int)((TAB_PER_B & 0xFFFF) << 16);         // tensor_dim0[15:0]
        g1[2] = (int)((TAB_PER_B >> 16) & 0xFFFF)          // tensor_dim0[31:16]
              | (1 << 16);                                 // tensor_dim1 = 1
        g1[3] = (int)((TAB_PER_B & 0xFFFF) << 16);         // tile_dim0 = 46800
        g1[4] = 0;                                         // tile_dim1/2 = 0
        g1[5] = (int)TAB_PER_B;                            // dim0_stride[31:0]
        g1[6] = 0;                                         // dim0_stride[47:32]
        g1[7] = 0;
        i32x4 z4 = {0, 0, 0, 0};
#if defined(__clang_major__) && (__clang_major__ >= 23)
        i32x8 z8 = {0, 0, 0, 0, 0, 0, 0, 0};
        __builtin_amdgcn_tensor_load_to_lds(g0, g1, z4, z4, z8, 0);
#else
        __builtin_amdgcn_tensor_load_to_lds(g0, g1, z4, z4, 0);
#endif
        __builtin_amdgcn_s_wait_tensorcnt(0);
    }
    // per-channel params -> LDS (all threads)
    for (int i = threadIdx.x; i < 256; i += 512) {
        lds[TAB_PER_B + i]       = ws[WS_AN + i];
        lds[TAB_PER_B + 256 + i] = ws[WS_BN + i];
        lds[TAB_PER_B + 512 + i] = cgb[i];
        lds[TAB_PER_B + 768 + i] = cbb[i];
    }
    __syncthreads();

    int wid = threadIdx.x >> 5, lane = threadIdx.x & 31;
    int h = (rt << 2) + (wid >> 2);
    int w = ((wid & 3) << 5) + lane;

    const int* segb = seg + ((size_t)b << 14);
    int og[9];
#pragma unroll
    for (int t = 0; t < 9; ++t) {
        int dy = t / 3 - 1, dx = t % 3 - 1;
        int hh = h + dy, ww = w + dx;
        int lab = 9;                                  // OOB tap -> zero row
        if (hh >= 0 && hh < 128 && ww >= 0 && ww < 128)
            lab = segb[(hh << 7) + ww];
        og[t] = (t * 10 + lab) * CPAD;
    }

    int pix = (h << 7) + w;
    const float* xb = x   + ((size_t)b << 22) + pix;
    float*       ob = out + ((size_t)b << 22) + pix;

    for (int c0 = 0; c0 < 256; c0 += 4) {
        v4f g  = *(const v4f*)&lds[TAB_PER_B + 512 + c0];   // conv biases seed
        v4f be = *(const v4f*)&lds[TAB_PER_B + 768 + c0];
#pragma unroll
        for (int t = 0; t < 9; ++t) {
            g  += *(const v4f*)&lds[og[t] + c0];
            be += *(const v4f*)&lds[og[t] + CONV_STRIDE + c0];
        }
        v4f aN = *(const v4f*)&lds[TAB_PER_B + c0];
        v4f bN = *(const v4f*)&lds[TAB_PER_B + 256 + c0];

        size_t xi = ((size_t)c0) << 14;
        float x0 = __builtin_nontemporal_load(xb + xi);
        float x1 = __builtin_nontemporal_load(xb + xi + 16384);
        float x2 = __builtin_nontemporal_load(xb + xi + 32768);
        float x3 = __builtin_nontemporal_load(xb + xi + 49152);
        float n0 = fmaf(aN.x, x0, bN.x);
        float n1 = fmaf(aN.y, x1, bN.y);
        float n2 = fmaf(aN.z, x2, bN.z);
        float n3 = fmaf(aN.w, x3, bN.w);
        __builtin_nontemporal_store(fmaf(n0, g.x, n0 + be.x), ob + xi);
        __builtin_nontemporal_store(fmaf(n1, g.y, n1 + be.y), ob + xi + 16384);
        __builtin_nontemporal_store(fmaf(n2, g.z, n2 + be.z), ob + xi + 32768);
        __builtin_nontemporal_store(fmaf(n3, g.w, n3 + be.w), ob + xi + 49152);
    }
}

extern "C" void kernel_launch(void* const* d_in, const int* in_sizes, int n_in,
                              void* d_out, int out_size, void* d_ws, size_t ws_size,
                              hipStream_t stream) {
    const float* x     = (const float*)d_in[0];
    const int*   seg   = (const int*)  d_in[1];
    const float* style = (const float*)d_in[2];
    const int*   exist = (const int*)  d_in[3];
    const float* bnw   = (const float*)d_in[4];
    const float* bnb   = (const float*)d_in[5];
    const float* fcw   = (const float*)d_in[6];
    const float* fcb   = (const float*)d_in[7];
    const float* cgw   = (const float*)d_in[8];
    const float* cgb   = (const float*)d_in[9];
    const float* cbw   = (const float*)d_in[10];
    const float* cbb   = (const float*)d_in[11];
    float* out = (float*)d_out;
    float* ws  = (float*)d_ws;

    // zero table (covers l=9 OOB row + c padding)
    hipMemsetAsync(ws + WS_GTAB, 0, (size_t)8 * TAB_PER_B * sizeof(float), stream);

    k_stats   <<<2048, 256, 0, stream>>>(x, ws);
    k_finstats<<<1,    256, 0, stream>>>(bnw, bnb, ws);
    k_mu      <<<32,   128, 0, stream>>>(fcw, fcb, style, exist, ws);
    k_tab     <<<360,  128, 0, stream>>>(cgw, cbw, ws);

    size_t ldsz = (size_t)(TAB_PER_B + 1024) * sizeof(float);  // ~191 KB (<320 KB/WGP)
    k_final   <<<256,  512, ldsz, stream>>>(x, seg, cgb, cbb, ws, out);
}